// SSO3Res_27487790694415
// MI455X (gfx1250) — compile-verified
//
#include <hip/hip_runtime.h>
#include <math.h>

typedef float v2f __attribute__((ext_vector_type(2)));
typedef float v8f __attribute__((ext_vector_type(8)));

#define TILE_M 128
#define TILE_N 128
#define TILE_K 16
#define A_STRIDE 20   // floats; 80B rows -> 16B aligned, conflict-free col reads
#define BN_STRIDE 132 // floats; 528B rows -> 16B aligned

// ---------------------------------------------------------------------------
// Batched fp32 GEMM using V_WMMA_F32_16X16X4_F32 with async global->LDS
// double-buffered pipeline (GLOBAL_LOAD_ASYNC_TO_LDS_B128 + s_wait_asynccnt).
//   C[m,n] = alpha * sum_k A[m,k]*B(k,n)  (+ bias[n]) (+ addm[m,n])
// BT=false: B stored row-major [K][N] (ldb = row stride)
// BT=true : B stored [N][K] (C = A @ B^T), kept untransposed in LDS
// Requirements: M % 128 == 0, K % 16 == 0 (true for all launches below).
// N may be ragged: async OOB loads are clamped to valid addresses; the
// garbage only feeds output columns that the epilogue never stores.
// ---------------------------------------------------------------------------
template <bool BT>
__global__ __launch_bounds__(256) void gemm_f32_wmma(
    const float* __restrict__ A, const float* __restrict__ B,
    float* __restrict__ C, const float* __restrict__ bias,
    const float* __restrict__ addm, int Mdim, int Ndim, int Kdim, int lda,
    int ldb, int ldc, long long bsA, long long bsB, long long bsC,
    long long bsAdd, float alpha) {
  __shared__ float As[2][TILE_M][A_STRIDE];
  __shared__ float Bsm[2][BT ? (TILE_N * A_STRIDE) : (TILE_K * BN_STRIDE)];

  const int batch = blockIdx.z;
  A += (long long)batch * bsA;
  B += (long long)batch * bsB;
  C += (long long)batch * bsC;
  const float* addp = addm ? addm + (long long)batch * bsAdd : nullptr;

  const int tid = threadIdx.x;
  const int lid = tid & 31;
  const int wave = tid >> 5;       // 8 waves
  const int wm = (wave >> 1) * 32; // 4 wave-rows
  const int wn = (wave & 1) * 64;  // 2 wave-cols
  const int rowBase = blockIdx.y * TILE_M;
  const int colBase = blockIdx.x * TILE_N;

  // per-thread copy-slice coordinates (2 x 512B float4 slices for A and B)
  const int a_row0 = tid >> 2;              // idx = tid
  const int a_row1 = (tid + 256) >> 2;      // idx = tid + 256
  const int a_k4 = (tid & 3) << 2;
  // BT: B tile is 128 rows x 16 k
  const int bt_n0 = a_row0, bt_n1 = a_row1, bt_k4 = a_k4;
  // BN: B tile is 16 k x 128 n
  const int bn_k0 = tid >> 5;
  const int bn_k1 = (tid + 256) >> 5;
  const int bn_n4 = (tid & 31) << 2;

  v8f acc[2][4] = {};

  // ---- async copy issue for one K-tile into buffer `buf` ----
  auto issue = [&](int kt, int buf) {
    {
      unsigned lo0 = (unsigned)(uintptr_t)&As[buf][a_row0][a_k4];
      const float* gp0 = A + (long long)(rowBase + a_row0) * lda + (kt + a_k4);
      asm volatile("global_load_async_to_lds_b128 %0, %1, off"
                   :: "v"(lo0), "v"(gp0) : "memory");
      unsigned lo1 = (unsigned)(uintptr_t)&As[buf][a_row1][a_k4];
      const float* gp1 = A + (long long)(rowBase + a_row1) * lda + (kt + a_k4);
      asm volatile("global_load_async_to_lds_b128 %0, %1, off"
                   :: "v"(lo1), "v"(gp1) : "memory");
    }
    if (BT) {
      int n0 = colBase + bt_n0; if (n0 > Ndim - 1) n0 = Ndim - 1;
      int n1 = colBase + bt_n1; if (n1 > Ndim - 1) n1 = Ndim - 1;
      unsigned lo0 = (unsigned)(uintptr_t)&Bsm[buf][bt_n0 * A_STRIDE + bt_k4];
      const float* gp0 = B + (long long)n0 * ldb + (kt + bt_k4);
      asm volatile("global_load_async_to_lds_b128 %0, %1, off"
                   :: "v"(lo0), "v"(gp0) : "memory");
      unsigned lo1 = (unsigned)(uintptr_t)&Bsm[buf][bt_n1 * A_STRIDE + bt_k4];
      const float* gp1 = B + (long long)n1 * ldb + (kt + bt_k4);
      asm volatile("global_load_async_to_lds_b128 %0, %1, off"
                   :: "v"(lo1), "v"(gp1) : "memory");
    } else {
      int c4 = colBase + bn_n4; if (c4 > Ndim - 4) c4 = Ndim - 4;
      unsigned lo0 = (unsigned)(uintptr_t)&Bsm[buf][bn_k0 * BN_STRIDE + bn_n4];
      const float* gp0 = B + (long long)(kt + bn_k0) * ldb + c4;
      asm volatile("global_load_async_to_lds_b128 %0, %1, off"
                   :: "v"(lo0), "v"(gp0) : "memory");
      unsigned lo1 = (unsigned)(uintptr_t)&Bsm[buf][bn_k1 * BN_STRIDE + bn_n4];
      const float* gp1 = B + (long long)(kt + bn_k1) * ldb + c4;
      asm volatile("global_load_async_to_lds_b128 %0, %1, off"
                   :: "v"(lo1), "v"(gp1) : "memory");
    }
  };

  const int nT = Kdim / TILE_K;
  issue(0, 0);

  const int ll = lid & 15;
  const int hl2 = (lid >> 4) << 1;

  for (int ti = 0; ti < nT; ++ti) {
    asm volatile("s_wait_asynccnt 0x0" ::: "memory");
    __syncthreads();
    if (ti + 1 < nT) issue((ti + 1) * TILE_K, (ti + 1) & 1);
    const int cur = ti & 1;

#pragma unroll
    for (int ks = 0; ks < TILE_K; ks += 4) {
      const int ak = ks + hl2;  // this lane-half's K pair {0,1} or {2,3}
      v2f af[2];
      af[0].x = As[cur][wm + ll][ak];
      af[0].y = As[cur][wm + ll][ak + 1];
      af[1].x = As[cur][wm + 16 + ll][ak];
      af[1].y = As[cur][wm + 16 + ll][ak + 1];
      v2f bf[4];
#pragma unroll
      for (int j = 0; j < 4; ++j) {
        const int col = wn + j * 16 + ll;
        if (BT) {
          bf[j].x = Bsm[cur][col * A_STRIDE + ak];
          bf[j].y = Bsm[cur][col * A_STRIDE + ak + 1];
        } else {
          bf[j].x = Bsm[cur][ak * BN_STRIDE + col];
          bf[j].y = Bsm[cur][(ak + 1) * BN_STRIDE + col];
        }
      }
#pragma unroll
      for (int i = 0; i < 2; ++i)
#pragma unroll
        for (int j = 0; j < 4; ++j)
          acc[i][j] = __builtin_amdgcn_wmma_f32_16x16x4_f32(
              false, af[i], false, bf[j], (short)0, acc[i][j], false, false);
    }
  }

  // ---- epilogue ----
  const int rowAdd = (lid >> 4) * 8;  // lanes 16-31 hold rows M = e+8
#pragma unroll
  for (int i = 0; i < 2; ++i) {
#pragma unroll
    for (int j = 0; j < 4; ++j) {
      int col = colBase + wn + j * 16 + ll;
      if (col < Ndim) {
#pragma unroll
        for (int e = 0; e < 8; ++e) {
          int row = rowBase + wm + i * 16 + rowAdd + e;
          long long off = (long long)row * ldc + col;
          float v = alpha * acc[i][j][e];
          if (bias) v += bias[col];
          if (addp) v += addp[off];
          C[off] = v;
        }
      }
    }
  }
}

// ---------------------------------------------------------------------------
// Radial MLP: 96 blocks, one per (t,g,b). r -> 32 -> LN -> relu -> 32 -> LN
// -> relu -> 10240.  R[t][g][b][o].
// ---------------------------------------------------------------------------
__global__ __launch_bounds__(256) void radial_mlp(
    const float* __restrict__ r, const float* __restrict__ W1,
    const float* __restrict__ b1, const float* __restrict__ g1,
    const float* __restrict__ be1, const float* __restrict__ W2,
    const float* __restrict__ b2, const float* __restrict__ g2,
    const float* __restrict__ be2, const float* __restrict__ W3,
    const float* __restrict__ b3, float* __restrict__ R) {
  const int t = blockIdx.x >> 5;
  const int rem = blockIdx.x & 31;
  const int g = rem >> 3;
  const int b = rem & 7;
  const int tg = t * 4 + g;
  __shared__ float h1[32], h2[32];
  __shared__ float mu_s, rs_s;
  const int tid = threadIdx.x;
  const float rb = r[b];

  if (tid < 32) h1[tid] = rb * W1[tg * 32 + tid] + b1[tg * 32 + tid];
  __syncthreads();
  if (tid == 0) {
    float mu = 0.f;
    for (int i = 0; i < 32; ++i) mu += h1[i];
    mu *= (1.f / 32.f);
    float var = 0.f;
    for (int i = 0; i < 32; ++i) {
      float d = h1[i] - mu;
      var += d * d;
    }
    mu_s = mu;
    rs_s = rsqrtf(var * (1.f / 32.f) + 1e-5f);
  }
  __syncthreads();
  if (tid < 32) {
    float v = (h1[tid] - mu_s) * rs_s * g1[tg * 32 + tid] + be1[tg * 32 + tid];
    h1[tid] = fmaxf(v, 0.f);
  }
  __syncthreads();
  if (tid < 32) {
    float a = b2[tg * 32 + tid];
    const float* w = W2 + (tg * 32 + tid) * 32;
    for (int i = 0; i < 32; ++i) a += h1[i] * w[i];
    h2[tid] = a;
  }
  __syncthreads();
  if (tid == 0) {
    float mu = 0.f;
    for (int i = 0; i < 32; ++i) mu += h2[i];
    mu *= (1.f / 32.f);
    float var = 0.f;
    for (int i = 0; i < 32; ++i) {
      float d = h2[i] - mu;
      var += d * d;
    }
    mu_s = mu;
    rs_s = rsqrtf(var * (1.f / 32.f) + 1e-5f);
  }
  __syncthreads();
  if (tid < 32) {
    float v = (h2[tid] - mu_s) * rs_s * g2[tg * 32 + tid] + be2[tg * 32 + tid];
    h2[tid] = fmaxf(v, 0.f);
  }
  __syncthreads();
  for (int o = tid; o < 10240; o += 256) {
    float a = b3[(long long)tg * 10240 + o];
    const float* w = W3 + ((long long)tg * 10240 + o) * 32;
#pragma unroll
    for (int i = 0; i < 32; ++i) a += h2[i] * w[i];
    R[((long long)tg * 8 + b) * 10240 + o] = a;
  }
}

// ---------------------------------------------------------------------------
// Materialize kern_g[t][b][f][k] = R[t,g,b,mo*64+mi] * basis_g[b,d] where the
// flat index f*Kg + k = mo*(dout*64) + d*64 + mi (faithful reshape view).
// ---------------------------------------------------------------------------
__global__ void kern_mat(float* __restrict__ out, const float* __restrict__ R,
                         const float* __restrict__ basisg, int g, int Kg,
                         int dout, int nTotal) {
  int idx = blockIdx.x * blockDim.x + threadIdx.x;
  if (idx >= nTotal) return;
  const int per_tb = 64 * Kg;
  int t = idx / (8 * per_tb);
  int r1 = idx - t * 8 * per_tb;
  int b = r1 / per_tb;
  int flat = r1 - b * per_tb;
  int mo = flat / (dout * 64);
  int rr = flat - mo * dout * 64;
  int d = rr >> 6;
  int mi = rr & 63;
  out[idx] = R[((long long)(t * 4 + g) * 8 + b) * 10240 + mo * 64 + mi] *
             basisg[b * dout + d];
}

// ---------------------------------------------------------------------------
// Column softmax over p axis of E[b][p][q]  (torch softmax dim=1 semantics).
// ---------------------------------------------------------------------------
__global__ __launch_bounds__(256) void softmax_col(float* __restrict__ E) {
  const int q = blockIdx.x * 256 + threadIdx.x;  // 0..1023, coalesced
  float* base = E + (long long)blockIdx.y * (1024 * 1024);
  float m = -3.402823466e38f;
  for (int p = 0; p < 1024; ++p) m = fmaxf(m, base[p * 1024 + q]);
  float s = 0.f;
  for (int p = 0; p < 1024; ++p) s += expf(base[p * 1024 + q] - m);
  const float inv = 1.f / s;
  for (int p = 0; p < 1024; ++p) {
    long long o = (long long)p * 1024 + q;
    base[o] = expf(base[o] - m) * inv;
  }
}

// ---------------------------------------------------------------------------
// Row LayerNorm + ReLU, in place.  One block per row of n elements.
// ---------------------------------------------------------------------------
__global__ __launch_bounds__(256) void ln_relu_rows(float* __restrict__ x,
                                                    const float* __restrict__ g,
                                                    const float* __restrict__ be,
                                                    int n) {
  __shared__ float red[256];
  __shared__ float mu_s, rs_s;
  const long long base = (long long)blockIdx.x * n;
  const int tid = threadIdx.x;
  float s = 0.f;
  for (int i = tid; i < n; i += 256) s += x[base + i];
  red[tid] = s;
  __syncthreads();
  for (int st = 128; st > 0; st >>= 1) {
    if (tid < st) red[tid] += red[tid + st];
    __syncthreads();
  }
  if (tid == 0) mu_s = red[0] / (float)n;
  __syncthreads();
  const float mu = mu_s;
  s = 0.f;
  for (int i = tid; i < n; i += 256) {
    float d = x[base + i] - mu;
    s += d * d;
  }
  red[tid] = s;
  __syncthreads();
  for (int st = 128; st > 0; st >>= 1) {
    if (tid < st) red[tid] += red[tid + st];
    __syncthreads();
  }
  if (tid == 0) rs_s = rsqrtf(red[0] / (float)n + 1e-5f);
  __syncthreads();
  const float rs = rs_s;
  for (int i = tid; i < n; i += 256) {
    float v = (x[base + i] - mu) * rs * g[i] + be[i];
    x[base + i] = fmaxf(v, 0.f);
  }
}

// ---------------------------------------------------------------------------
static void launch_gemm(bool bt, const float* A, const float* B, float* C,
                        const float* bias, const float* addm, int M, int N,
                        int K, int lda, int ldb, int ldc, long long bsA,
                        long long bsB, long long bsC, long long bsAdd,
                        int batch, float alpha, hipStream_t stream) {
  dim3 grid((N + TILE_N - 1) / TILE_N, M / TILE_M, batch);
  dim3 blk(256);
  if (bt)
    gemm_f32_wmma<true><<<grid, blk, 0, stream>>>(A, B, C, bias, addm, M, N, K,
                                                  lda, ldb, ldc, bsA, bsB, bsC,
                                                  bsAdd, alpha);
  else
    gemm_f32_wmma<false><<<grid, blk, 0, stream>>>(A, B, C, bias, addm, M, N, K,
                                                   lda, ldb, ldc, bsA, bsB, bsC,
                                                   bsAdd, alpha);
}

extern "C" void kernel_launch(void* const* d_in, const int* in_sizes, int n_in,
                              void* d_out, int out_size, void* d_ws,
                              size_t ws_size, hipStream_t stream) {
  (void)in_sizes; (void)n_in; (void)out_size; (void)ws_size;

  const float* features = (const float*)d_in[0];   // [8,1024,64]
  const float* r        = (const float*)d_in[1];   // [8,1]
  const float* basisp[4] = {(const float*)d_in[2], (const float*)d_in[3],
                            (const float*)d_in[4], (const float*)d_in[5]};
  const float* rf_W1  = (const float*)d_in[6];
  const float* rf_b1  = (const float*)d_in[7];
  const float* rf_g1  = (const float*)d_in[8];
  const float* rf_be1 = (const float*)d_in[9];
  const float* rf_W2  = (const float*)d_in[10];
  const float* rf_b2  = (const float*)d_in[11];
  const float* rf_g2  = (const float*)d_in[12];
  const float* rf_be2 = (const float*)d_in[13];
  const float* rf_W3  = (const float*)d_in[14];
  const float* rf_b3  = (const float*)d_in[15];
  const float* transform = (const float*)d_in[16]; // [8,64,5120]
  const float* Wl1 = (const float*)d_in[17];       // [5120,2560]
  const float* bl1 = (const float*)d_in[18];
  const float* ln_g = (const float*)d_in[19];
  const float* ln_b = (const float*)d_in[20];
  const float* Wl2 = (const float*)d_in[21];       // [5120,5120]
  const float* bl2 = (const float*)d_in[22];
  float* out = (float*)d_out;                      // [8,1024,5120]

  // workspace layout (floats)
  float* ws = (float*)d_ws;
  float* Rbuf = ws;                         //  983,040  [3,4,8,10240]
  float* kern = Rbuf + 983040;              //  3,932,160 (all g concatenated)
  float* qkvg = kern + 3932160;             //  27,525,120 (max over g)
  float* E    = qkvg + 27525120;            //  8,388,608  [8,1024,1024]
  float* z    = E + 8388608;                //  20,971,520 [8,1024,2560]
  float* zl   = z + 20971520;               //  41,943,040 [8192,5120]

  // 1) radial MLPs -> R
  radial_mlp<<<96, 256, 0, stream>>>(r, rf_W1, rf_b1, rf_g1, rf_be1, rf_W2,
                                     rf_b2, rf_g2, rf_be2, rf_W3, rf_b3, Rbuf);

  const int offs[4] = {0, 160, 640, 1440};
  for (int g = 0; g < 4; ++g) {
    const int dout = 2 * g + 1;
    const int Kg = 160 * dout;
    float* kerng = kern + 1536LL * offs[g];  // [3][8][64][Kg]
    const int nTot = 1536 * Kg;

    // 2) materialize faithful-view kernel
    kern_mat<<<(nTot + 255) / 256, 256, 0, stream>>>(kerng, Rbuf, basisp[g], g,
                                                     Kg, dout, nTot);
    // 3) qkv = features @ kern, per branch t (batch over b)
    for (int t = 0; t < 3; ++t)
      launch_gemm(false, features, kerng + (long long)t * 512 * Kg,
                  qkvg + (long long)t * 8192 * Kg, nullptr, nullptr, 1024, Kg,
                  64, 64, Kg, Kg, 65536LL, 64LL * Kg, 1024LL * Kg, 0, 8, 1.f,
                  stream);
    // 4) E = q @ k^T / sqrt(Kg)
    launch_gemm(true, qkvg + 2LL * 8192 * Kg, qkvg + 1LL * 8192 * Kg, E,
                nullptr, nullptr, 1024, 1024, Kg, Kg, Kg, 1024, 1024LL * Kg,
                1024LL * Kg, 1048576LL, 0, 8, 1.f / sqrtf((float)Kg), stream);
    // 5) softmax over p axis (column-normalized)
    softmax_col<<<dim3(4, 8), 256, 0, stream>>>(E);
    // 6) z[:, :, off:off+Kg] = alpha @ v
    launch_gemm(false, E, qkvg, z + offs[g], nullptr, nullptr, 1024, Kg, 1024,
                1024, Kg, 2560, 1048576LL, 1024LL * Kg, 1024LL * 2560, 0, 8,
                1.f, stream);
  }

  // 7) proj -> d_out (read back in final GEMM epilogue)
  launch_gemm(false, features, transform, out, nullptr, nullptr, 1024, 5120, 64,
              64, 5120, 5120, 65536LL, 327680LL, 5242880LL, 0, 8, 1.f, stream);
  // 8) zl = z @ Wl1^T + bl1   (8192 x 5120 x 2560)
  launch_gemm(true, z, Wl1, zl, bl1, nullptr, 8192, 5120, 2560, 2560, 2560,
              5120, 0, 0, 0, 0, 1, 1.f, stream);
  // 9) LayerNorm + ReLU in place
  ln_relu_rows<<<8192, 256, 0, stream>>>(zl, ln_g, ln_b, 5120);
  // 10) out = zl @ Wl2^T + bl2 + proj   (8192 x 5120 x 5120)
  launch_gemm(true, zl, Wl2, out, bl2, out, 8192, 5120, 5120, 5120, 5120, 5120,
              0, 0, 0, 0, 1, 1.f, stream);
}